// FinePreprocess_12412455485716
// MI455X (gfx1250) — compile-verified
//
#include <hip/hip_runtime.h>
#include <hip/hip_bf16.h>

typedef __attribute__((ext_vector_type(2))) float v2f;
typedef __attribute__((ext_vector_type(8))) float v8f;

#define WKS 5
#define K2 25
#define CTR 12
#define NUMK 8
#define EPSV 1e-8f
#define CF 128
#define HF 240
#define WF 320
#define DC 256
#define WCC 80
#define LC 4800

// ---------------------------------------------------------------------------
// Kernel 1: per-match patch gather + cosine sims + top-k + sel scatter + fc
// One block (128 threads) per match.
// ---------------------------------------------------------------------------
__global__ __launch_bounds__(128) void fine_prep_kernel(
    const float* __restrict__ f0, const float* __restrict__ f1,
    const float* __restrict__ c0f, const float* __restrict__ c1f,
    const float* __restrict__ Wd, const float* __restrict__ bd,
    const int* __restrict__ bi, const int* __restrict__ ii, const int* __restrict__ ji,
    float* __restrict__ sel0w, float* __restrict__ sel1w,
    float* __restrict__ fc0w, float* __restrict__ fc1w,
    float* __restrict__ posOut)
{
    __shared__ float p0[K2 * CF];
    __shared__ float p1[K2 * CF];
    __shared__ float xr0[DC], xr1[DC];
    __shared__ float d0s[K2], n0s[K2], d1s[K2], n1s[K2];
    __shared__ float s0s[K2 - 1], s1s[K2];
    __shared__ int   i0s[NUMK], i1s[NUMK];

    const int m  = blockIdx.x;
    const int tid = threadIdx.x;           // == channel, 0..127
    const int b  = bi[m];
    const int ip = ii[m];
    const int jp = ji[m];

    // Stage feat_c rows (independent of patches)
    for (int k = tid; k < DC; k += 128) {
        xr0[k] = c0f[((size_t)b * LC + ip) * DC + k];
        xr1[k] = c1f[((size_t)b * LC + jp) * DC + k];
    }

    // Gather 5x5 patch, image 0 (channel = tid; 5 contiguous floats per row)
    {
        const int hc = ip / WCC, wc = ip % WCC;
        const int h0 = hc * 4 - 2, w0 = wc * 4 - 2;
        for (int r = 0; r < WKS; ++r) {
            const int h = h0 + r;
            const bool hv = (h >= 0) && (h < HF);
            const float* src = f0 + (((size_t)b * CF + tid) * HF + h) * WF;
            float* dst = &p0[(r * WKS) * CF + tid];
            for (int c = 0; c < WKS; ++c) {
                const int w = w0 + c;
                float v = 0.0f;
                if (hv && w >= 0 && w < WF) v = src[w];
                dst[c * CF] = v;
            }
        }
    }
    // Gather 5x5 patch, image 1
    {
        const int hc = jp / WCC, wc = jp % WCC;
        const int h0 = hc * 4 - 2, w0 = wc * 4 - 2;
        for (int r = 0; r < WKS; ++r) {
            const int h = h0 + r;
            const bool hv = (h >= 0) && (h < HF);
            const float* src = f1 + (((size_t)b * CF + tid) * HF + h) * WF;
            float* dst = &p1[(r * WKS) * CF + tid];
            for (int c = 0; c < WKS; ++c) {
                const int w = w0 + c;
                float v = 0.0f;
                if (hv && w >= 0 && w < WF) v = src[w];
                dst[c * CF] = v;
            }
        }
    }
    __syncthreads();

    // Dots with center + squared norms.
    // Rotate channel start per thread to spread LDS bank accesses
    // (bank = (cc + p*8) % 64 instead of all threads hitting ch%64).
    if (tid < K2) {
        const int rot = (tid * 8) & (CF - 1);
        float d = 0.f, n = 0.f;
        for (int cc = 0; cc < CF; ++cc) {
            const int ch = (cc + rot) & (CF - 1);
            const float cv = p0[CTR * CF + ch];
            const float v  = p0[tid * CF + ch];
            d += cv * v; n += v * v;
        }
        d0s[tid] = d; n0s[tid] = n;
    } else if (tid >= 64 && tid < 64 + K2) {
        const int pp = tid - 64;
        const int rot = (pp * 8) & (CF - 1);
        float d = 0.f, n = 0.f;
        for (int cc = 0; cc < CF; ++cc) {
            const int ch = (cc + rot) & (CF - 1);
            const float cv = p0[CTR * CF + ch];
            const float v  = p1[pp * CF + ch];
            d += cv * v; n += v * v;
        }
        d1s[pp] = d; n1s[pp] = n;
    }
    __syncthreads();

    // Cosine similarities
    {
        const float nc = fmaxf(sqrtf(n0s[CTR]), EPSV);
        if (tid < K2 - 1) {
            const int pz = (tid < CTR) ? tid : tid + 1;   // skip center
            s0s[tid] = d0s[pz] / (nc * fmaxf(sqrtf(n0s[pz]), EPSV));
        } else if (tid >= 64 && tid < 64 + K2) {
            const int pp = tid - 64;
            s1s[pp] = d1s[pp] / (nc * fmaxf(sqrtf(n1s[pp]), EPSV));
        }
    }
    __syncthreads();

    // Stable top-k (ties -> lowest index, matching jax.lax.top_k)
    if (tid == 0) {
        unsigned used = 0u;
        i0s[0] = CTR;  // reference prepends ctr, then uses sim0 indices directly on full patch
        for (int t = 0; t < NUMK - 1; ++t) {
            float best = -3.402823466e38f; int bidx = 0;
            for (int k = 0; k < K2 - 1; ++k)
                if (!((used >> k) & 1u) && s0s[k] > best) { best = s0s[k]; bidx = k; }
            used |= 1u << bidx;
            i0s[t + 1] = bidx;
        }
        used = 0u;
        for (int t = 0; t < NUMK; ++t) {
            float best = -3.402823466e38f; int bidx = 0;
            for (int k = 0; k < K2; ++k)
                if (!((used >> k) & 1u) && s1s[k] > best) { best = s1s[k]; bidx = k; }
            used |= 1u << bidx;
            i1s[t] = bidx;
        }
    }
    __syncthreads();

    // pos1 (written as float values)
    if (tid < NUMK) {
        const int iv = i1s[tid];
        posOut[((size_t)m * NUMK + tid) * 2 + 0] = (float)(iv / WKS);
        posOut[((size_t)m * NUMK + tid) * 2 + 1] = (float)(iv % WKS);
    }

    // Scatter selected rows to workspace
    for (int kk = 0; kk < NUMK; ++kk) {
        sel0w[((size_t)m * NUMK + kk) * CF + tid] = p0[i0s[kk] * CF + tid];
        sel1w[((size_t)m * NUMK + kk) * CF + tid] = p1[i1s[kk] * CF + tid];
    }

    // fc = xc @ W_down + b_down  (coalesced W_down reads across lanes;
    // xr reads are wave-uniform broadcasts -> no bank conflicts)
    {
        float a0 = bd[tid], a1 = bd[tid];
        for (int k = 0; k < DC; ++k) {
            const float w = Wd[k * CF + tid];
            a0 += xr0[k] * w;
            a1 += xr1[k] * w;
        }
        fc0w[(size_t)m * CF + tid] = a0;
        fc1w[(size_t)m * CF + tid] = a1;
    }
}

// ---------------------------------------------------------------------------
// Kernel 2: fp32 WMMA GEMM  Y(rows x 128) = A(rows x 128) @ W(128 x 128)
//           [+ bias[col]] [+ addv[(row/8)*128 + col]]
// 256 threads = 8 waves; block computes a 16x128 output tile, each wave a
// 16x16 sub-tile via V_WMMA_F32_16X16X4_F32. The 16x128 A tile is one
// contiguous 8KB block -> staged once into LDS (coalesced float4 loads),
// eliminating the 8x redundant A fetch across waves. B (W_merge half, 64KB)
// is shared by every block and served from WGP$/L2.
// Fragment layouts per CDNA5 ISA 7.12.2:
//   A 16x4:  lanes 0-15 M=lane (K=k0,k0+1), lanes 16-31 M=lane-16 (K=k0+2,k0+3)
//   B 4x16:  lanes 0-15 N=lane (K=k0,k0+1), lanes 16-31 N=lane-16 (K=k0+2,k0+3)
//   C/D 16x16: vgpr v, lanes 0-15 -> M=v, lanes 16-31 -> M=v+8; N = lane&15
// ---------------------------------------------------------------------------
__global__ __launch_bounds__(256) void merge_gemm_kernel(
    const float* __restrict__ A,
    const float* __restrict__ W,
    const float* __restrict__ bias, const float* __restrict__ addv,
    float* __restrict__ Y, int rows)
{
    __shared__ float atile[16 * CF];   // 8 KB

    const int lane = threadIdx.x & 31;
    const int wv   = threadIdx.x >> 5;
    const int row0 = blockIdx.x * 16;
    const int n    = wv * 16 + (lane & 15);       // output column
    const int koff = (lane >> 4) << 1;            // 0 or 2

    // Stage contiguous 16x128 A tile into LDS (2 float4 per thread)
    {
        const float* tsrc = A + (size_t)row0 * CF;
        const int validElems = (rows - row0) * CF;   // >= 2048 except tail
        if (validElems >= 16 * CF) {
            const float4* s4 = (const float4*)tsrc;
            float4* d4 = (float4*)atile;
            #pragma unroll
            for (int e4 = threadIdx.x; e4 < (16 * CF) / 4; e4 += 256)
                d4[e4] = s4[e4];
        } else {
            for (int e = threadIdx.x; e < 16 * CF; e += 256)
                atile[e] = (e < validElems) ? tsrc[e] : 0.0f;
        }
    }
    __syncthreads();

    const float* arow = atile + (lane & 15) * CF + koff;

    v8f acc = {0.f, 0.f, 0.f, 0.f, 0.f, 0.f, 0.f, 0.f};
    #pragma unroll 8
    for (int k0 = 0; k0 < CF; k0 += 4) {
        v2f av; av.x = arow[k0]; av.y = arow[k0 + 1];          // ds_load_b64
        const float* bp = W + (size_t)(k0 + koff) * CF + n;
        v2f bv; bv.x = bp[0]; bv.y = bp[CF];
        acc = __builtin_amdgcn_wmma_f32_16x16x4_f32(
            /*neg_a=*/false, av, /*neg_b=*/false, bv,
            /*c_mod=*/(short)0, acc, /*reuse_a=*/false, /*reuse_b=*/false);
    }

    const int rbase = row0 + ((lane >> 4) << 3);
    const float badd = bias ? bias[n] : 0.0f;
    #pragma unroll
    for (int v = 0; v < 8; ++v) {
        const int r = rbase + v;
        if (r < rows) {
            float val = acc[v] + badd;
            if (addv) val += addv[(size_t)(r >> 3) * CF + n];
            Y[(size_t)r * CF + n] = val;
        }
    }
}

// ---------------------------------------------------------------------------
extern "C" void kernel_launch(void* const* d_in, const int* in_sizes, int n_in,
                              void* d_out, int out_size, void* d_ws, size_t ws_size,
                              hipStream_t stream)
{
    (void)n_in; (void)out_size; (void)ws_size;

    const float* f0  = (const float*)d_in[0];
    const float* f1  = (const float*)d_in[1];
    const float* c0f = (const float*)d_in[2];
    const float* c1f = (const float*)d_in[3];
    const float* Wd  = (const float*)d_in[4];
    const float* bd  = (const float*)d_in[5];
    const float* Wm  = (const float*)d_in[6];
    const float* bm  = (const float*)d_in[7];
    const int*   bi  = (const int*)d_in[8];
    const int*   ii  = (const int*)d_in[9];
    const int*   ji  = (const int*)d_in[10];
    const int M = in_sizes[8];

    // workspace layout (floats)
    float* ws    = (float*)d_ws;
    float* sel0w = ws;                               // M*8*128
    float* sel1w = sel0w + (size_t)M * NUMK * CF;    // M*8*128
    float* fc0w  = sel1w + (size_t)M * NUMK * CF;    // M*128
    float* fc1w  = fc0w + (size_t)M * CF;            // M*128
    float* g0w   = fc1w + (size_t)M * CF;            // M*128
    float* g1w   = g0w  + (size_t)M * CF;            // M*128

    // output layout: sel0 | sel1 | pos1 (flattened, return order)
    float* out     = (float*)d_out;
    float* outSel0 = out;
    float* outSel1 = out + (size_t)M * NUMK * CF;
    float* outPos  = outSel1 + (size_t)M * NUMK * CF;

    fine_prep_kernel<<<M, 128, 0, stream>>>(
        f0, f1, c0f, c1f, Wd, bd, bi, ii, ji,
        sel0w, sel1w, fc0w, fc1w, outPos);

    const float* WmTop = Wm;            // rows 0..127 of W_merge (sel part)
    const float* WmBot = Wm + CF * CF;  // rows 128..255 of W_merge (fc part)

    // g = fc @ Wm_bot  (identical for all 8 rows of a match -> computed once)
    const int gbFc = (M + 15) / 16;
    merge_gemm_kernel<<<gbFc, 256, 0, stream>>>(fc0w, WmBot, nullptr, nullptr, g0w, M);
    merge_gemm_kernel<<<gbFc, 256, 0, stream>>>(fc1w, WmBot, nullptr, nullptr, g1w, M);

    // out = sel_raw @ Wm_top + g[row/8] + b_merge
    const int rowsSel = M * NUMK;
    const int gbSel = (rowsSel + 15) / 16;
    merge_gemm_kernel<<<gbSel, 256, 0, stream>>>(sel0w, WmTop, bm, g0w, outSel0, rowsSel);
    merge_gemm_kernel<<<gbSel, 256, 0, stream>>>(sel1w, WmTop, bm, g1w, outSel1, rowsSel);
}